// TokenEmbedding_51153060495497
// MI455X (gfx1250) — compile-verified
//
#include <hip/hip_runtime.h>

// CDNA5 (gfx1250) wave32 WMMA implementation.
//
// Roofline: 269 MB of output writes vs 23.3 TB/s HBM -> ~11.5 us floor.
// 1.07 GFLOP fp32 via V_WMMA_F32_16X16X4_F32 (exact f32, K=8 done as 2xK=4).
// GEMM orientation: M = kernel index k (74 pad 80, 5 tiles), N = (b,t,c)
// output columns (16 per wave), K = tap index j. D-fragment: each lane holds
// 8 consecutive k values of one column -> contiguous output stores.

typedef __attribute__((ext_vector_type(2))) float v2f;
typedef __attribute__((ext_vector_type(8))) float v8f;

#define BATCH 256
#define LSEQ  512
#define CCH   7
#define TPOS  513          // L + 1 window positions
#define NK    74
#define KS    8
#define PAD   4
#define COLS_PER_BATCH (TPOS * CCH)          // 3591
#define TOTAL_COLS     (BATCH * COLS_PER_BATCH)  // 919296 = 16 * 57456
#define OUT_STRIDE_T   512
#define OUT_PER_BATCH  (TPOS * OUT_STRIDE_T)

__global__ __launch_bounds__(256)
void conv_taps_wmma_kernel(const float* __restrict__ x,
                           const float* __restrict__ kernels,
                           float* __restrict__ out)
{
    const int lane = threadIdx.x & 31;
    const int wave = (int)((blockIdx.x * blockDim.x + threadIdx.x) >> 5);
    const int n    = lane & 15;        // column slot within tile / A row M
    const int h    = lane >> 4;        // lane half: selects K pair {2h,2h+1}

    const int col = wave * 16 + n;     // global output column (b,t,c)
    if (col >= TOTAL_COLS) return;     // exact launch; guard is free

    int b = col / COLS_PER_BATCH;
    int u = col - b * COLS_PER_BATCH;
    int t = u / CCH;
    int c = u - t * CCH;

    // ---- B fragment: win[(t,c)][j], j in {2h,2h+1} (B1) and {4+2h,4+2h+1} (B2)
    // B 4x16 f32 layout: VGPRs hold K pairs; lanes 0-15 K={0,1}, lanes 16-31 K={2,3}.
    const float* xb = x + ((long long)b * LSEQ) * CCH + c;
    v2f B1, B2;
    {
        int j0 = 2 * h;
        int xi;
        xi = t + j0 + 0 - PAD; B1.x = (xi >= 0 && xi < LSEQ) ? xb[xi * CCH] : 0.0f;
        xi = t + j0 + 1 - PAD; B1.y = (xi >= 0 && xi < LSEQ) ? xb[xi * CCH] : 0.0f;
        xi = t + j0 + 4 - PAD; B2.x = (xi >= 0 && xi < LSEQ) ? xb[xi * CCH] : 0.0f;
        xi = t + j0 + 5 - PAD; B2.y = (xi >= 0 && xi < LSEQ) ? xb[xi * CCH] : 0.0f;
    }

    const long long obase = (long long)b * OUT_PER_BATCH + (long long)t * OUT_STRIDE_T;
    const int m_lo = h * 8;            // D rows held by this lane half

    #pragma unroll
    for (int mt = 0; mt < 5; ++mt) {
        // ---- A fragment: keff[k][j] = kernels[k][j][1], k = mt*16 + n.
        // A 16x4 f32 layout: lanes 0-15 M rows with K={0,1}, lanes 16-31 K={2,3}.
        const int krow = mt * 16 + n;
        v2f A1 = {0.0f, 0.0f}, A2 = {0.0f, 0.0f};
        if (krow < NK) {
            const float* kb = kernels + (long long)krow * (KS * 3) + 1; // column 1
            int j0 = 2 * h;
            A1.x = kb[(j0 + 0) * 3];
            A1.y = kb[(j0 + 1) * 3];
            A2.x = kb[(j0 + 4) * 3];
            A2.y = kb[(j0 + 5) * 3];
        }

        v8f acc = {};
        acc = __builtin_amdgcn_wmma_f32_16x16x4_f32(
                  false, A1, false, B1, (short)0, acc, false, false);
        acc = __builtin_amdgcn_wmma_f32_16x16x4_f32(
                  false, A2, false, B2, (short)0, acc, false, false);

        // ---- Store: lane's column is (b,t,c); rows k = mt*16 + m_lo + e,
        // contiguous in output at c*73 + k. Output (269MB) > L2 (192MB):
        // use non-temporal stores for the pure write stream.
        #pragma unroll
        for (int e = 0; e < 8; ++e) {
            const int k = mt * 16 + m_lo + e;
            const float v = acc[e];
            if (k < NK - 1) {
                __builtin_nontemporal_store(v, out + obase + c * (NK - 1) + k);
            } else if (k == NK - 1 && c == 0) {
                __builtin_nontemporal_store(v, out + obase + (OUT_STRIDE_T - 1));
            }
        }
    }
}

extern "C" void kernel_launch(void* const* d_in, const int* in_sizes, int n_in,
                              void* d_out, int out_size, void* d_ws, size_t ws_size,
                              hipStream_t stream) {
    const float* x       = (const float*)d_in[0];   // (256, 512, 7) f32
    const float* kernels = (const float*)d_in[1];   // (74, 8, 3)  f32
    float* out = (float*)d_out;                     // 256*512*513 f32

    const int total_waves     = TOTAL_COLS / 16;    // 57456, exact
    const int waves_per_block = 8;                  // 256 threads, wave32
    const int blocks          = total_waves / waves_per_block; // 7182, exact

    conv_taps_wmma_kernel<<<blocks, 256, 0, stream>>>(x, kernels, out);
}